// BiGeaR_tch_7516192768529
// MI455X (gfx1250) — compile-verified
//
#include <hip/hip_runtime.h>
#include <hip/hip_bf16.h>
#include <math.h>

#define DIM 64

typedef __attribute__((ext_vector_type(2))) float v2f;
typedef __attribute__((ext_vector_type(8))) float v8f;

// Force the native global_atomic_add_f32 (no CAS retry loop). The SpMM is
// atomic-throughput bound (768M f32 adds into L2), so this lowering matters.
__device__ __forceinline__ void atomAddF32(float* p, float v) {
#if defined(__HIP_DEVICE_COMPILE__)
  unsafeAtomicAdd(p, v);
#else
  atomicAdd(p, v);
#endif
}

// ---------------------------------------------------------------- utilities

__global__ void zero_out_kernel(float* out) {
  if (threadIdx.x < 2 && blockIdx.x == 0) out[threadIdx.x] = 0.0f;
}

__global__ void concat_kernel(const float* __restrict__ uw,
                              const float* __restrict__ iw,
                              float* __restrict__ x, long uN, long total) {
  long t = (long)blockIdx.x * blockDim.x + threadIdx.x;
  if (t >= total) return;
  x[t] = (t < uN) ? uw[t] : iw[t - uN];
}

__global__ void zero_buf_kernel(float* __restrict__ p, long n,
                                int layer, const int* __restrict__ nl) {
  if (layer >= *nl) return;
  long t = (long)blockIdx.x * blockDim.x + threadIdx.x;
  if (t < n) p[t] = 0.0f;
}

// ------------------------------------------------ COO SpMM scatter (L2 atomics)
// 16 threads per edge; each thread gathers a float4 of x[col] and scatters
// 4 native f32 atomic adds into x_next[row]. x (38.4 MB) is L2-resident
// (192 MB L2), so both the gather and the scatter stay on-chip; HBM only
// streams the 48 MB/layer of edge data.

__global__ void spmm_scatter_kernel(const float* __restrict__ x,
                                    float* __restrict__ y,
                                    const float* __restrict__ vals,
                                    const int* __restrict__ rows,
                                    const int* __restrict__ cols,
                                    long E, int layer, const int* __restrict__ nl) {
  if (layer >= *nl) return;
  long t = (long)blockIdx.x * blockDim.x + threadIdx.x;
  long e = t >> 4;
  if (e >= E) return;
  int q = (int)(t & 15) * 4;
  // stream-ahead hint for the edge arrays (lowers to global_prefetch_b8)
  if (e + 4096 < E) {
    __builtin_prefetch(&vals[e + 4096], 0, 0);
    __builtin_prefetch(&cols[e + 4096], 0, 0);
  }
  int c = cols[e];
  int r = rows[e];
  float w = vals[e];
  const float4 v = *(const float4*)(x + (long)c * DIM + q);
  float* dst = y + (long)r * DIM + q;
  atomAddF32(dst + 0, w * v.x);
  atomAddF32(dst + 1, w * v.y);
  atomAddF32(dst + 2, w * v.z);
  atomAddF32(dst + 3, w * v.w);
}

// --------------------------------- accumulate sampled rows across layer outputs

__global__ void accum_sampled_kernel(const float* __restrict__ x,
                                     const int* __restrict__ ui,
                                     const int* __restrict__ pi,
                                     const int* __restrict__ ni,
                                     float* __restrict__ ua,
                                     float* __restrict__ pa,
                                     float* __restrict__ na,
                                     int B, long num_users,
                                     int mode, int layer, const int* __restrict__ nl) {
  if (mode == 1 && layer >= *nl) return;
  long t = (long)blockIdx.x * blockDim.x + threadIdx.x;
  int i = (int)(t >> 4);
  if (i >= B) return;
  int q = (int)(t & 15) * 4;
  long ur = (long)ui[i];
  long pr = num_users + (long)pi[i];
  long nr = num_users + (long)ni[i];
  const float4 vu = *(const float4*)(x + ur * DIM + q);
  const float4 vp = *(const float4*)(x + pr * DIM + q);
  const float4 vn = *(const float4*)(x + nr * DIM + q);
  float4* du = (float4*)(ua + (long)i * DIM + q);
  float4* dp = (float4*)(pa + (long)i * DIM + q);
  float4* dn = (float4*)(na + (long)i * DIM + q);
  if (mode == 0) {
    *du = vu; *dp = vp; *dn = vn;
  } else {
    float4 a = *du; a.x += vu.x; a.y += vu.y; a.z += vu.z; a.w += vu.w; *du = a;
    float4 b = *dp; b.x += vp.x; b.y += vp.y; b.z += vp.z; b.w += vp.w; *dp = b;
    float4 cc = *dn; cc.x += vn.x; cc.y += vn.y; cc.z += vn.z; cc.w += vn.w; *dn = cc;
  }
}

// ------------------------------------------------- WMMA f32 batched dot + loss
// One wave per 16 batch rows. D = U(16x64) x P^T(64x16) via 16 chained
// V_WMMA_F32_16X16X4_F32; the diagonal holds the per-row dot products.
// A layout (16x4 f32): lane -> M = lane&15, VGPRs r -> K = 2*(lane>>4)+r.
// B layout (4x16 f32): lane -> N = lane&15, same K slicing -> identical loads.
// Diag (i,i): i<8 at lane i, VGPR i; i>=8 at lane i+16, VGPR i-8.

__device__ __forceinline__ float pick8(v8f v, int i) {
  float r = v[0];
#pragma unroll
  for (int k = 1; k < 8; ++k)
    if (i == k) r = v[k];
  return r;
}

__global__ void wmma_loss_kernel(const float* __restrict__ ua,
                                 const float* __restrict__ pa,
                                 const float* __restrict__ na,
                                 float* __restrict__ out,
                                 int B, const int* __restrict__ nl) {
  int wave = (int)((blockIdx.x * blockDim.x + threadIdx.x) >> 5);
  int lane = (int)(threadIdx.x & 31);
  int r0 = wave * 16;
  if (r0 >= B) return;                    // wave-uniform: EXEC stays all-1s
  int row = r0 + (lane & 15);
  int kh = (lane >> 4) << 1;              // K sub-slice base: 0 or 2
  const float* up = ua + (long)row * DIM;
  const float* pp = pa + (long)row * DIM;
  const float* np = na + (long)row * DIM;

  float invL = 1.0f / (float)(*nl + 1);   // pooled mean scale
  float scale = invL * invL;              // applied to dot of two pooled vecs

#if __has_builtin(__builtin_amdgcn_wmma_f32_16x16x4_f32)
  v8f cp = {};
  v8f cn = {};
#pragma unroll
  for (int c = 0; c < 16; ++c) {
    int kb = c * 4 + kh;
    v2f a  = *(const v2f*)(up + kb);
    v2f bp = *(const v2f*)(pp + kb);
    v2f bn = *(const v2f*)(np + kb);
    cp = __builtin_amdgcn_wmma_f32_16x16x4_f32(false, a, false, bp, (short)0, cp, false, false);
    cn = __builtin_amdgcn_wmma_f32_16x16x4_f32(false, a, false, bn, (short)0, cn, false, false);
  }
  bool valid = (lane < 8) || (lane >= 24);
  int idx = (lane < 8) ? lane : ((lane >= 24) ? (lane - 24) : 0);
  float sp = pick8(cp, idx) * scale;
  float sn = pick8(cn, idx) * scale;
#else
  // scalar fallback: lanes 0..15 each reduce one row; lanes 16..31 idle
  bool valid = (lane < 16);
  float dp = 0.0f, dn = 0.0f;
  for (int k = 0; k < DIM; ++k) {
    dp += up[k] * pp[k];
    dn += up[k] * np[k];
  }
  float sp = dp * scale;
  float sn = dn * scale;
#endif
  float xv = sn - sp;
  // numerically stable softplus
  float spl = fmaxf(xv, 0.0f) + log1pf(expf(-fabsf(xv)));
  if (valid) atomAddF32(out, spl / (float)B);
}

// ----------------------------------------------------------- L2 regularization

__global__ void reg_loss_kernel(const float* __restrict__ uw,
                                const float* __restrict__ iw,
                                const int* __restrict__ ui,
                                const int* __restrict__ pi,
                                const int* __restrict__ ni,
                                float* __restrict__ out, int B) {
  long t = (long)blockIdx.x * blockDim.x + threadIdx.x;
  int i = (int)(t >> 4);
  if (i >= B) return;
  int q = (int)(t & 15) * 4;
  const float4 a = *(const float4*)(uw + (long)ui[i] * DIM + q);
  const float4 b = *(const float4*)(iw + (long)pi[i] * DIM + q);
  const float4 c = *(const float4*)(iw + (long)ni[i] * DIM + q);
  float s = a.x * a.x + a.y * a.y + a.z * a.z + a.w * a.w
          + b.x * b.x + b.y * b.y + b.z * b.z + b.w * b.w
          + c.x * c.x + c.y * c.y + c.z * c.z + c.w * c.w;
  atomAddF32(out + 1, 0.5f * s / (float)B);
}

// ------------------------------------------------------------------- launcher

static inline unsigned nblocks(long n, int t) {
  return (unsigned)((n + (long)t - 1) / (long)t);
}

extern "C" void kernel_launch(void* const* d_in, const int* in_sizes, int n_in,
                              void* d_out, int out_size, void* d_ws, size_t ws_size,
                              hipStream_t stream) {
  const float* uw   = (const float*)d_in[0];   // [NUM_USERS, 64]
  const float* iw   = (const float*)d_in[1];   // [NUM_ITEMS, 64]
  const float* vals = (const float*)d_in[2];   // [E]
  const int*   rows = (const int*)d_in[3];     // [E]
  const int*   cols = (const int*)d_in[4];     // [E]
  const int*   ui   = (const int*)d_in[5];     // [B]
  const int*   pi   = (const int*)d_in[6];     // [B]
  const int*   ni   = (const int*)d_in[7];     // [B]
  const int*   nl   = (const int*)d_in[8];     // scalar num_layers (device)

  long uN = (long)in_sizes[0];                 // NUM_USERS*64
  long iN = (long)in_sizes[1];                 // NUM_ITEMS*64
  long E  = (long)in_sizes[2];
  int  B  = in_sizes[5];
  long num_users = uN / DIM;
  long NV = uN + iN;                           // N * DIM

  float* ws     = (float*)d_ws;
  float* x_cur  = ws;                          // N*64
  float* x_next = ws + NV;                     // N*64
  float* ua     = ws + 2 * NV;                 // B*64
  float* pa     = ua + (long)B * DIM;          // B*64
  float* na     = pa + (long)B * DIM;          // B*64
  float* out    = (float*)d_out;

  const int T = 256;

  zero_out_kernel<<<1, 32, 0, stream>>>(out);
  concat_kernel<<<nblocks(NV, T), T, 0, stream>>>(uw, iw, x_cur, uN, NV);
  // layer-0 contribution to pooled embeddings at sampled rows
  accum_sampled_kernel<<<nblocks((long)B * 16, T), T, 0, stream>>>(
      x_cur, ui, pi, ni, ua, pa, na, B, num_users, /*mode=*/0, /*layer=*/0, nl);

  for (int layer = 0; layer < 3; ++layer) {
    zero_buf_kernel<<<nblocks(NV, T), T, 0, stream>>>(x_next, NV, layer, nl);
    spmm_scatter_kernel<<<nblocks(E * 16, T), T, 0, stream>>>(
        x_cur, x_next, vals, rows, cols, E, layer, nl);
    accum_sampled_kernel<<<nblocks((long)B * 16, T), T, 0, stream>>>(
        x_next, ui, pi, ni, ua, pa, na, B, num_users, /*mode=*/1, layer, nl);
    float* tmp = x_cur; x_cur = x_next; x_next = tmp;
  }

  // 512 tiles of 16 rows -> 512 waves -> 64 blocks of 256 threads
  long waves = ((long)B + 15) / 16;
  wmma_loss_kernel<<<nblocks(waves * 32, T), T, 0, stream>>>(ua, pa, na, out, B, nl);
  reg_loss_kernel<<<nblocks((long)B * 16, T), T, 0, stream>>>(uw, iw, ui, pi, ni, out, B);
}